// ModulePSBS_36094905156103
// MI455X (gfx1250) — compile-verified
//
#include <hip/hip_runtime.h>
#include <cstdint>

// ---------------------------------------------------------------------------
// Clements mesh forward (B=512, N=256, L=256, A=128), fp32 complex.
// One batch row per wave32; 8 modes (16 VGPRs) per lane. Layer parameters are
// double-buffered in LDS via gfx1250 async global->LDS copies (ASYNCcnt).
// ---------------------------------------------------------------------------

#define LAYERS 256
#define NMODES 256
#define BATCH  512

__device__ __forceinline__ unsigned lds_byte_off(const void* p) {
  // low 32 bits of a flat shared-space address == workgroup-relative LDS offset
  return (unsigned)(uintptr_t)p;
}

// Issue the 6 async copies staging one layer's params (theta, split, atten_re,
// atten_im) into this wave's LDS stage buffer. 16B per lane per instruction.
// s_wait_dscnt 0 first: guarantee prior ds reads of the target buffer retired.
__device__ __forceinline__ void issue_layer_async(
    const float* th_p, const float* sp_p, const float* are_p, const float* aim_p,
    int l, int lane, unsigned smbase) {
  unsigned offp = (unsigned)((l * 128 + 4 * lane) * 4);   // bytes into (L,A)
  unsigned offa = (unsigned)((l * 256 + 8 * lane) * 4);   // bytes into (L,N)
  unsigned d_th = smbase + 16u * (unsigned)lane;          // [   0, 512)
  unsigned d_sp = smbase + 512u + 16u * (unsigned)lane;   // [ 512,1024)
  unsigned d_ar = smbase + 1024u + 32u * (unsigned)lane;  // [1024,2048)
  unsigned d_ai = smbase + 2048u + 32u * (unsigned)lane;  // [2048,3072)
  asm volatile(
      "s_wait_dscnt 0x0\n\t"
      "global_load_async_to_lds_b128 %0, %4, %6\n\t"
      "global_load_async_to_lds_b128 %1, %4, %7\n\t"
      "global_load_async_to_lds_b128 %2, %5, %8\n\t"
      "global_load_async_to_lds_b128 %2, %5, %8 offset:16\n\t"
      "global_load_async_to_lds_b128 %3, %5, %9\n\t"
      "global_load_async_to_lds_b128 %3, %5, %9 offset:16"
      :
      : "v"(d_th), "v"(d_sp), "v"(d_ar), "v"(d_ai),
        "v"(offp), "v"(offa),
        "s"(th_p), "s"(sp_p), "s"(are_p), "s"(aim_p)
      : "memory");
}

__device__ __forceinline__ void wait_async_le6() {
  asm volatile("s_wait_asynccnt 0x6" ::: "memory");
}
__device__ __forceinline__ void wait_async_0() {
  asm volatile("s_wait_asynccnt 0x0" ::: "memory");
}

// 2x2 photonic mixer:  na = c*e*a + i*sa*b ;  nb = i*sa*e*a + c*b
// with c=cos(pi/4+sp), sa=sin(pi/4+sp), e=exp(i*theta)=(cs,sn).
__device__ __forceinline__ void mix2(float& ar, float& ai, float& br, float& bi,
                                     float cs, float sn, float ca, float sa) {
  const float cer = ca * cs, cei = ca * sn;     // c*e
  const float ser = -sa * sn, sei = sa * cs;    // i*sa*e
  const float nar = cer * ar - cei * ai - sa * bi;
  const float nai = cer * ai + cei * ar + sa * br;
  const float nbr = ser * ar - sei * ai + ca * br;
  const float nbi = ser * ai + sei * ar + ca * bi;
  ar = nar; ai = nai; br = nbr; bi = nbi;
}

__global__ __launch_bounds__(32) void psbs_mesh_kernel(
    const float* __restrict__ x_real, const float* __restrict__ x_imag,
    const float* __restrict__ theta_p, const float* __restrict__ split_p,
    const float* __restrict__ atre_p, const float* __restrict__ atim_p,
    float* __restrict__ out) {
  __shared__ __align__(16) float smem[2][768];  // 2 stages x 3KB (one wave/block)

  const int lane  = threadIdx.x;        // 0..31
  const int row   = blockIdx.x;         // 0..511 (one batch row per wave)
  const int mbase = 8 * lane;           // this lane owns modes mbase..mbase+7

  // ---- load state: 8 complex modes per lane --------------------------------
  float xr[8], xi[8];
  {
    const float4* r4 = (const float4*)(x_real + row * NMODES + mbase);
    const float4* i4 = (const float4*)(x_imag + row * NMODES + mbase);
    float4 a = r4[0], b = r4[1], c = i4[0], d = i4[1];
    xr[0]=a.x; xr[1]=a.y; xr[2]=a.z; xr[3]=a.w; xr[4]=b.x; xr[5]=b.y; xr[6]=b.z; xr[7]=b.w;
    xi[0]=c.x; xi[1]=c.y; xi[2]=c.z; xi[3]=c.w; xi[4]=d.x; xi[5]=d.y; xi[6]=d.z; xi[7]=d.w;
  }

  const unsigned smb0 = lds_byte_off(&smem[0][0]);
  const unsigned smb1 = lds_byte_off(&smem[1][0]);

  // prologue: stage layer 0 into buffer 0
  issue_layer_async(theta_p, split_p, atre_p, atim_p, 0, lane, smb0);

  const float PI4 = 0.78539816339744830962f;

  for (int l = 0; l < LAYERS; ++l) {
    const int buf = l & 1;
    if (l + 1 < LAYERS) {
      issue_layer_async(theta_p, split_p, atre_p, atim_p, l + 1, lane,
                        ((l + 1) & 1) ? smb1 : smb0);
      wait_async_le6();   // 12 outstanding -> retire current layer's 6
    } else {
      wait_async_0();     // final layer: drain
    }

    const float* sm = &smem[buf][0];
    const float4 th4 = *(const float4*)(sm + 4 * lane);
    const float4 sp4 = *(const float4*)(sm + 128 + 4 * lane);
    const float4 ar0 = *(const float4*)(sm + 256 + 8 * lane);
    const float4 ar1 = *(const float4*)(sm + 256 + 8 * lane + 4);
    const float4 ai0 = *(const float4*)(sm + 512 + 8 * lane);
    const float4 ai1 = *(const float4*)(sm + 512 + 8 * lane + 4);

    // ---- attenuation: x *= atten (complex, per mode) -----------------------
    const float are[8] = {ar0.x, ar0.y, ar0.z, ar0.w, ar1.x, ar1.y, ar1.z, ar1.w};
    const float aim[8] = {ai0.x, ai0.y, ai0.z, ai0.w, ai1.x, ai1.y, ai1.z, ai1.w};
#pragma unroll
    for (int j = 0; j < 8; ++j) {
      const float r = xr[j] * are[j] - xi[j] * aim[j];
      const float i = xr[j] * aim[j] + xi[j] * are[j];
      xr[j] = r; xi[j] = i;
    }

    // ---- per-pair coefficients: this lane owns pairs 4*lane .. 4*lane+3 ----
    const float th[4] = {th4.x, th4.y, th4.z, th4.w};
    const float spl[4] = {sp4.x, sp4.y, sp4.z, sp4.w};
    float sn[4], cs[4], sa[4], ca[4];
#pragma unroll
    for (int t = 0; t < 4; ++t) {
      __sincosf(th[t], &sn[t], &cs[t]);
      __sincosf(PI4 + spl[t], &sa[t], &ca[t]);
    }

    if ((l & 1) == 0) {
      // even layer: pairs (2a, 2a+1) -> fully in-lane: (0,1)(2,3)(4,5)(6,7)
#pragma unroll
      for (int t = 0; t < 4; ++t)
        mix2(xr[2 * t], xi[2 * t], xr[2 * t + 1], xi[2 * t + 1],
             cs[t], sn[t], ca[t], sa[t]);
    } else {
      // odd layer: pairs (2a+1, 2a+2); local (1,2)(3,4)(5,6) in-lane,
      // pair a=4*lane+3 straddles the lane boundary (modes 8k+7, 8k+8).
      // Capture neighbor state/coeffs before any update.
      const float nx0r = __shfl_down(xr[0], 1);  // next lane's mode 8(k+1)
      const float nx0i = __shfl_down(xi[0], 1);
      const float px7r = __shfl_up(xr[7], 1);    // prev lane's mode 8k-1
      const float px7i = __shfl_up(xi[7], 1);
      const float psn = __shfl_up(sn[3], 1);     // prev lane's boundary coeffs
      const float pcs = __shfl_up(cs[3], 1);
      const float psa = __shfl_up(sa[3], 1);
      const float pca = __shfl_up(ca[3], 1);

#pragma unroll
      for (int t = 0; t < 3; ++t)
        mix2(xr[2 * t + 1], xi[2 * t + 1], xr[2 * t + 2], xi[2 * t + 2],
             cs[t], sn[t], ca[t], sa[t]);

      if (lane < 31) {  // 'a' side of boundary pair (mode 255 passes through)
        const float cer = ca[3] * cs[3], cei = ca[3] * sn[3];
        const float ar_ = xr[7], ai_ = xi[7];
        xr[7] = cer * ar_ - cei * ai_ - sa[3] * nx0i;
        xi[7] = cer * ai_ + cei * ar_ + sa[3] * nx0r;
      }
      if (lane > 0) {   // 'b' side of boundary pair (mode 0 passes through)
        const float ser = -psa * psn, sei = psa * pcs;
        const float ar_ = px7r, ai_ = px7i;
        const float br = xr[0], bi = xi[0];
        xr[0] = ser * ar_ - sei * ai_ + pca * br;
        xi[0] = ser * ai_ + sei * ar_ + pca * bi;
      }
    }
  }

  // ---- store (B, N, 2): lane's 8 modes = 16 contiguous floats --------------
  float4* o = (float4*)(out + row * NMODES * 2 + 16 * lane);
  o[0] = make_float4(xr[0], xi[0], xr[1], xi[1]);
  o[1] = make_float4(xr[2], xi[2], xr[3], xi[3]);
  o[2] = make_float4(xr[4], xi[4], xr[5], xi[5]);
  o[3] = make_float4(xr[6], xi[6], xr[7], xi[7]);
}

extern "C" void kernel_launch(void* const* d_in, const int* in_sizes, int n_in,
                              void* d_out, int out_size, void* d_ws, size_t ws_size,
                              hipStream_t stream) {
  (void)in_sizes; (void)n_in; (void)out_size; (void)d_ws; (void)ws_size;
  const float* x_real = (const float*)d_in[0];
  const float* x_imag = (const float*)d_in[1];
  const float* theta  = (const float*)d_in[2];
  // d_in[3] = index (L, N/2, 2) int32 — deterministic Clements pattern, derived
  // analytically from layer parity in the kernel.
  const float* split  = (const float*)d_in[4];
  const float* at_re  = (const float*)d_in[5];
  const float* at_im  = (const float*)d_in[6];
  float* out = (float*)d_out;

  psbs_mesh_kernel<<<dim3(BATCH), dim3(32), 0, stream>>>(
      x_real, x_imag, theta, split, at_re, at_im, out);
}